// NoiseModel_41180146434211
// MI455X (gfx1250) — compile-verified
//
#include <hip/hip_runtime.h>

#define NPIX   4194304      // 4*4*512*512
#define HALFN  2097152      // NPIX / 2 (threefry counter pairing)
#define NCAT   101
#define NBINS  101
#define CDFSZ  (NBINS * NCAT)   // 10201 floats = 40804 B, fits in LDS
#define MPAD   112              // 101 padded up to 7 WMMA tiles of 16

typedef __attribute__((ext_vector_type(2))) float v2f;
typedef __attribute__((ext_vector_type(8))) float v8f;
typedef __attribute__((ext_vector_type(4))) int   v4i;

#if defined(__has_builtin)
#if __has_builtin(__builtin_amdgcn_global_load_async_to_lds_b128) && \
    __has_builtin(__builtin_amdgcn_s_wait_asynccnt)
#define HAVE_ASYNC_LDS 1
#endif
#endif
#ifndef HAVE_ASYNC_LDS
#define HAVE_ASYNC_LDS 0
#endif

// ---------------------------------------------------------------------------
// Kernel A: cdf = noise_matrix (101x101) x U (101x101), U[k][j] = (k <= j).
// One wave32 per 16x16 output tile, V_WMMA_F32_16X16X4_F32, K-loop of 28.
// B operand is synthesized from the predicate (no memory). Edges zero-padded;
// EXEC reconverges to all-ones before every WMMA.
// ---------------------------------------------------------------------------
__global__ __launch_bounds__(32) void cdf_wmma_kernel(const float* __restrict__ nm,
                                                      float* __restrict__ cdf) {
    const int tile = blockIdx.x;          // 0..48
    const int tm   = (tile / 7) * 16;     // output row tile
    const int tn   = (tile % 7) * 16;     // output col tile
    const int lane = threadIdx.x & 31;
    const int hi   = lane >> 4;           // lane half (K split for A/B operands)
    const int l    = lane & 15;

    const int row  = tm + l;              // A row handled by this lane
    const int col  = tn + l;              // B/D column handled by this lane
    const int rowc = row < NBINS ? row : NBINS - 1;

    v8f c = {};
    for (int k0 = 0; k0 < MPAD; k0 += 4) {
        const int ka   = k0 + 2 * hi;                 // this lane's K base
        const int ka0c = (ka     < NCAT) ? ka     : NCAT - 1;
        const int ka1c = (ka + 1 < NCAT) ? ka + 1 : NCAT - 1;
        const float a0 = nm[rowc * NCAT + ka0c];
        const float a1 = nm[rowc * NCAT + ka1c];
        v2f a, b;
        a.x = (row < NBINS && ka     < NCAT) ? a0 : 0.0f;
        a.y = (row < NBINS && ka + 1 < NCAT) ? a1 : 0.0f;
        // upper-triangular ones matrix, generated on the fly
        b.x = (ka     <= col) ? 1.0f : 0.0f;
        b.y = (ka + 1 <= col) ? 1.0f : 0.0f;
        c = __builtin_amdgcn_wmma_f32_16x16x4_f32(false, a, false, b,
                                                  (short)0, c, false, false);
    }
    // C/D layout: VGPR r -> M = r (lanes 0-15) / r+8 (lanes 16-31), N = lane%16
#pragma unroll
    for (int r = 0; r < 8; ++r) {
        const int orow = tm + r + 8 * hi;
        if (orow < NBINS && col < NCAT) cdf[orow * NCAT + col] = c[r];
    }
}

// ---------------------------------------------------------------------------
// threefry2x32-20, key = (0, 1)  (jax.random.key(1))
// ---------------------------------------------------------------------------
__device__ __forceinline__ unsigned rotl32(unsigned v, int n) {
    return (v << n) | (v >> (32 - n));
}

__device__ __forceinline__ void threefry2x32(unsigned x0, unsigned x1,
                                             unsigned& o0, unsigned& o1) {
    const unsigned ks1 = 1u, ks2 = 0x1BD11BDBu;   // 0 ^ 1 ^ 0x1BD11BDA
    x1 += ks1;                                    // x0 += ks0 (= 0)
#define TFR(r) { x0 += x1; x1 = rotl32(x1, r) ^ x0; }
    TFR(13) TFR(15) TFR(26) TFR(6)   x0 += ks1; x1 += ks2 + 1u;
    TFR(17) TFR(29) TFR(16) TFR(24)  x0 += ks2; x1 += 2u;
    TFR(13) TFR(15) TFR(26) TFR(6)              x1 += ks1 + 3u;
    TFR(17) TFR(29) TFR(16) TFR(24)  x0 += ks1; x1 += ks2 + 4u;
    TFR(13) TFR(15) TFR(26) TFR(6)   x0 += ks2; x1 += 5u;
#undef TFR
    o0 = x0; o1 = x1;
}

// branchless binary search: idx = first j with cdf_row[j] >= u, clipped to 100
__device__ __forceinline__ float sample_one(float xv, float u,
                                            const float* __restrict__ s_cdf) {
    int r = (int)(xv * 200.0f) + 1;               // digitize(x, arange(0,.5,.005))
    r = r > NBINS - 1 ? NBINS - 1 : r;
    r = r < 0 ? 0 : r;
    const int rb = r * NCAT;
    int idx = 0;
#pragma unroll
    for (int s = 64; s >= 1; s >>= 1) {
        const int   cand = idx + s;
        const int   cc   = cand > NCAT ? NCAT : cand;   // clamp LDS address
        const float v    = s_cdf[rb + cc - 1];
        idx = (cand <= NCAT && v < u) ? cand : idx;     // v_cndmask, no branch
    }
    idx = idx > NCAT - 1 ? NCAT - 1 : idx;
    return fmaf(0.0002f, (float)idx, -0.0101f);         // noisy_ranges[idx]
}

// ---------------------------------------------------------------------------
// Kernel B: 8 pixels/thread (pairs p, p+N/2 share one threefry call),
// float4 global I/O, CDF staged in LDS — via async global->LDS b128 (ASYNCcnt)
// when the toolchain exposes the gfx1250 builtin, else load+ds_store_b128.
// ---------------------------------------------------------------------------
__global__ __launch_bounds__(256) void noise_sample_kernel(const float* __restrict__ x,
                                                           const float* __restrict__ cdf,
                                                           float* __restrict__ out) {
    __shared__ __align__(16) float s_cdf[CDFSZ];
#if HAVE_ASYNC_LDS
    {
        for (int i = threadIdx.x; i < CDFSZ / 4; i += 256) {
            __builtin_amdgcn_global_load_async_to_lds_b128(
                (__attribute__((address_space(1))) v4i*)(cdf + 4 * i),
                (__attribute__((address_space(3))) v4i*)(&s_cdf[4 * i]),
                0, 0);
        }
        if (threadIdx.x == 0) s_cdf[CDFSZ - 1] = cdf[CDFSZ - 1];
        __builtin_amdgcn_s_wait_asynccnt(0);
    }
#else
    {
        const float4* src = (const float4*)cdf;
        for (int i = threadIdx.x; i < CDFSZ / 4; i += 256) {
            const float4 v = src[i];
            *(float4*)&s_cdf[4 * i] = v;                 // ds_store_b128
        }
        if (threadIdx.x == 0) s_cdf[CDFSZ - 1] = cdf[CDFSZ - 1];
    }
#endif
    __syncthreads();

    const unsigned t    = blockIdx.x * 256u + threadIdx.x;   // 0..524287
    const unsigned base = t * 4u;                            // < HALFN
    const float4 xa = *(const float4*)(x + base);
    const float4 xb = *(const float4*)(x + base + HALFN);
    const float xav[4] = {xa.x, xa.y, xa.z, xa.w};
    const float xbv[4] = {xb.x, xb.y, xb.z, xb.w};
    float ra[4], rbv[4];
#pragma unroll
    for (int j = 0; j < 4; ++j) {
        unsigned b0, b1;
        threefry2x32(base + j, base + j + HALFN, b0, b1);    // 2 uniforms / call
        const float u0 = __uint_as_float((b0 >> 9) | 0x3f800000u) - 1.0f;
        const float u1 = __uint_as_float((b1 >> 9) | 0x3f800000u) - 1.0f;
        ra[j]  = sample_one(xav[j], u0, s_cdf);
        rbv[j] = sample_one(xbv[j], u1, s_cdf);
    }
    *(float4*)(out + base)         = make_float4(ra[0], ra[1], ra[2], ra[3]);
    *(float4*)(out + base + HALFN) = make_float4(rbv[0], rbv[1], rbv[2], rbv[3]);
}

extern "C" void kernel_launch(void* const* d_in, const int* in_sizes, int n_in,
                              void* d_out, int out_size, void* d_ws, size_t ws_size,
                              hipStream_t stream) {
    const float* x  = (const float*)d_in[0];   // (4,4,512,512) f32
    const float* nm = (const float*)d_in[1];   // (101,101) f32
    float* out = (float*)d_out;                // (4,4,512,512) f32
    float* cdf = (float*)d_ws;                 // 10201 f32 scratch

    hipLaunchKernelGGL(cdf_wmma_kernel, dim3(49), dim3(32), 0, stream, nm, cdf);
    hipLaunchKernelGGL(noise_sample_kernel, dim3(NPIX / (8 * 256)), dim3(256), 0, stream,
                       x, cdf, out);
}